// CorticalInhibitoryNetwork_36893769072821
// MI455X (gfx1250) — compile-verified
//
#include <hip/hip_runtime.h>

// ---------------------------------------------------------------------------
// CDNA5 (gfx1250) cortical inhibitory network step.
// Bandwidth-bound GEMV core: V_WMMA_F32_16X16X4_F32 + Tensor Data Mover
// (tensor_load_to_lds) for weight-tile staging.
// ---------------------------------------------------------------------------

typedef __attribute__((ext_vector_type(2))) float v2f;
typedef __attribute__((ext_vector_type(8))) float v8f;
typedef __attribute__((ext_vector_type(4))) unsigned int v4u;
typedef __attribute__((ext_vector_type(8))) int v8i;
typedef __attribute__((ext_vector_type(4))) int v4i;

#define PYR 30000
#define PV  3000
#define SST 2250
#define VIP 1500
#define NGC 750

#define KC    512      // K-chunk staged in LDS per iteration
#define WAVES 8        // 256 threads = 8 wave32
#define KSLICE (KC / WAVES)   // 64 K per wave per chunk -> 16 WMMAs

// workspace layout (floats)
#define WS_VIP 0
#define WS_NGC 1536
#define WS_GAP 2304

#if defined(__has_builtin)
#if __has_builtin(__builtin_amdgcn_tensor_load_to_lds)
#define HAVE_TDM 1
#endif
#endif
#ifndef HAVE_TDM
#define HAVE_TDM 0
#endif

// Guarded, zero-padded manual staging path (K tails / no-TDM fallback).
__device__ __forceinline__ void stage_tile_manual(float* sW, const float* __restrict__ W,
                                                  int row0, int k0, int M, int K, int tid)
{
    for (int f = tid * 4; f < 16 * KC; f += 256 * 4) {
        int r  = f / KC;
        int c  = f - r * KC;
        int gr = row0 + r;
        int gk = k0 + c;
        float4 v = make_float4(0.f, 0.f, 0.f, 0.f);
        if (gr < M) {
            const float* p = W + (size_t)gr * (size_t)K + gk;
            if (gk + 3 < K) {
                v = *(const float4*)p;
            } else {
                if (gk + 0 < K) v.x = p[0];
                if (gk + 1 < K) v.y = p[1];
                if (gk + 2 < K) v.z = p[2];
            }
        }
        *(float4*)&sW[f] = v;
    }
}

// ---------------------------------------------------------------------------
// Fused GEMV kernel: block ranges pick {w_lr_vip, w_lr_ngc, w_pv_gap}.
// Each block computes a 16-row tile of y = W*x (and, for PV, the fused
// gap term y - prev * rowsum(W)).
// ---------------------------------------------------------------------------
__global__ __launch_bounds__(256) void gemv_wmma_kernel(
    const float* __restrict__ Wvip,
    const float* __restrict__ Wngc,
    const float* __restrict__ Wgap,
    const float* __restrict__ xlr,       // long_range_excitation [PYR]
    const float* __restrict__ prev_pv,   // prev_pv_mem [PV]
    float* __restrict__ ws,
    int vipTiles, int ngcTiles)
{
    __shared__ float sW[16 * KC];          // 32 KB weight tile
    __shared__ float sX[KC];               // 2 KB vector chunk
    __shared__ float red[WAVES][16];       // per-wave partial y
    __shared__ float rsm[WAVES * 32];      // per-lane rowsum partials (PV only)

    const float* W;
    const float* x;
    float* y;
    int M, K;
    bool doGap = false;

    int bid = (int)blockIdx.x;
    if (bid < vipTiles) {
        W = Wvip; x = xlr; y = ws + WS_VIP; M = VIP; K = PYR;
    } else if (bid < vipTiles + ngcTiles) {
        bid -= vipTiles;
        W = Wngc; x = xlr; y = ws + WS_NGC; M = NGC; K = PYR;
    } else {
        bid -= vipTiles + ngcTiles;
        W = Wgap; x = prev_pv; y = ws + WS_GAP; M = PV; K = PV;
        doGap = true;
    }

    const int row0 = bid * 16;
    const int tid  = (int)threadIdx.x;
    const int lane = tid & 31;
    const int wave = tid >> 5;
    const int m    = lane & 15;     // A-fragment row owned by this lane
    const int half = lane >> 4;     // lane-half selects K pair {0,1} vs {2,3}
    const int validRows = (M - row0 < 16) ? (M - row0) : 16;

    v8f  acc = {};                  // 16x16 f32 accumulator (all columns equal)
    float rs = 0.0f;                // per-lane rowsum partial (PV only)

    for (int k0 = 0; k0 < K; k0 += KC) {
        const int kc = (K - k0 < KC) ? (K - k0) : KC;

        // ---- stage x chunk (coalesced, zero-padded tail) ----
        for (int i = tid; i < KC; i += 256) {
            int kg = k0 + i;
            sX[i] = (kg < K) ? x[kg] : 0.0f;
        }

#if HAVE_TDM
        if (kc == KC) {
            // rows beyond M: zero-fill manually (TDM writes only validRows rows)
            if (validRows < 16) {
                for (int f = validRows * KC + tid; f < 16 * KC; f += 256) sW[f] = 0.0f;
            }
            if (wave == 0) {
                // --- Tensor DMA: 16 x 512 f32 tile, row stride K, into sW ---
                unsigned long long ga = (unsigned long long)(size_t)W +
                    ((unsigned long long)(unsigned)row0 * (unsigned)K + (unsigned)k0) * 4ull;
                unsigned int ldsa = (unsigned int)(size_t)&sW[0];   // LDS offset = addr[31:0]
                unsigned int td0  = (unsigned int)(K - k0);         // extent from tile origin
                unsigned int td1  = (unsigned int)(M - row0);
                unsigned long long st0 = (unsigned long long)(unsigned)K;

                v4u g0 = { 1u,                                       // count=1 (valid, user)
                           ldsa,                                     // lds_addr
                           (unsigned int)(ga & 0xFFFFFFFFull),       // global_addr[31:0]
                           (unsigned int)((ga >> 32) & 0x1FFFFFFull) // global_addr[56:32]
                               | (2u << 30) };                       // type=2 (image)
                v8i g1 = { (int)(2u << 16),                          // data_size=4B; mask=0
                           (int)((td0 & 0xFFFFu) << 16),             // tensor_dim0[15:0]
                           (int)((td0 >> 16) | ((td1 & 0xFFFFu) << 16)),
                           (int)((td1 >> 16) | ((unsigned)KC << 16)),// tile_dim0=512
                           (int)(unsigned)validRows,                 // tile_dim1; tile_dim2=0
                           (int)(unsigned)(st0 & 0xFFFFFFFFull),     // dim0_stride[31:0]
                           (int)(unsigned)((st0 >> 32) & 0xFFFFull), // dim0_stride[47:32]
                           0 };                                      // dim1_stride (2D: unused)
                v4i g2 = { 0, 0, 0, 0 };
                v4i g3 = { 0, 0, 0, 0 };
                v8i g4 = { 0, 0, 0, 0, 0, 0, 0, 0 };                 // extended group (unused)
                __builtin_amdgcn_tensor_load_to_lds(g0, g1, g2, g3, g4, 0);
                __builtin_amdgcn_s_wait_tensorcnt(0);
            }
        } else {
            stage_tile_manual(sW, W, row0, k0, M, K, tid);
        }
#else
        stage_tile_manual(sW, W, row0, k0, M, K, tid);
#endif
        __syncthreads();

        // ---- this wave's 64-wide K slice: 16 chained WMMAs ----
        const int kkbase = wave * KSLICE;
        #pragma unroll
        for (int kk = 0; kk < KSLICE; kk += 4) {
            const int ks = kkbase + kk + half * 2;
            // A fragment: lane m holds W[row0+m][ks], W[row0+m][ks+1]
            v2f a = *(const v2f*)&sW[m * KC + ks];
            // B fragment: x chunk replicated across all 16 columns
            v2f b = *(const v2f*)&sX[ks];
            rs += a.x + a.y;  // fused rowsum (free: data already in VGPRs)
            acc = __builtin_amdgcn_wmma_f32_16x16x4_f32(
                /*neg_a=*/false, a, /*neg_b=*/false, b,
                /*c_mod=*/(short)0, acc, /*reuse_a=*/false, /*reuse_b=*/false);
        }
        __syncthreads();
    }

    // ---- cross-wave reduction ----
    // D layout: VGPR j, lanes 0-15 -> row j ; lanes 16-31 -> row j+8.
    // All columns equal, so lane 0 holds y[0..7], lane 16 holds y[8..15].
    if (lane == 0 || lane == 16) {
        const int base = half * 8;
        #pragma unroll
        for (int j = 0; j < 8; ++j) red[wave][base + j] = acc[j];
    }
    if (doGap) rsm[wave * 32 + lane] = rs;
    __syncthreads();

    if (tid < 16) {
        float s = 0.0f;
        #pragma unroll
        for (int w = 0; w < WAVES; ++w) s += red[w][tid];
        const int gr = row0 + tid;
        if (gr < M) {
            if (doGap) {
                float rstot = 0.0f;
                #pragma unroll
                for (int w = 0; w < WAVES; ++w)
                    rstot += rsm[w * 32 + tid] + rsm[w * 32 + tid + 16];
                y[gr] = s - prev_pv[gr] * rstot;   // gap-junction term, fused
            } else {
                y[gr] = s;
            }
        }
    }
}

// ---------------------------------------------------------------------------
// Elementwise LIF update for all four populations (7500 neurons total).
// ---------------------------------------------------------------------------
__device__ __forceinline__ void lif_step(float v, float ad, float gE, float gI,
                                         float extra, float gL, float tau,
                                         float vth, float& spk, float& mem)
{
    const float C  = gL * tau;
    const float I  = gL * (0.0f - v) + gE * (3.0f - v) + gI * (-0.5f - v) + extra;
    const float vn = v + 0.5f * I / C;
    const float s  = (vn >= vth + ad) ? 1.0f : 0.0f;
    spk = s;
    mem = (s > 0.0f) ? 0.0f : vn;
}

__global__ __launch_bounds__(256) void lif_kernel(
    const float* __restrict__ pv_g_exc,  const float* __restrict__ pv_g_inh,
    const float* __restrict__ sst_g_exc, const float* __restrict__ sst_g_inh,
    const float* __restrict__ vip_g_exc, const float* __restrict__ vip_g_inh,
    const float* __restrict__ ngc_g_exc, const float* __restrict__ ngc_g_inh,
    const float* __restrict__ ffwd,      const float* __restrict__ ach,
    const float* __restrict__ nic_tr,    const float* __restrict__ mus_tr,
    const float* __restrict__ pv_gL,  const float* __restrict__ pv_tau,
    const float* __restrict__ pv_vth, const float* __restrict__ pv_v,
    const float* __restrict__ pv_ad,
    const float* __restrict__ sst_gL,  const float* __restrict__ sst_tau,
    const float* __restrict__ sst_vth, const float* __restrict__ sst_v,
    const float* __restrict__ sst_ad,
    const float* __restrict__ vip_gL,  const float* __restrict__ vip_tau,
    const float* __restrict__ vip_vth, const float* __restrict__ vip_v,
    const float* __restrict__ vip_ad,
    const float* __restrict__ ngc_gL,  const float* __restrict__ ngc_tau,
    const float* __restrict__ ngc_vth, const float* __restrict__ ngc_v,
    const float* __restrict__ ngc_ad,
    const float* __restrict__ ws,
    float* __restrict__ out)
{
    const int i = (int)blockIdx.x * 256 + (int)threadIdx.x;
    const float a = ach[0];
    // output offsets (spk then mem, concatenated in return order)
    if (i < PV) {
        const int j = i;
        float spk, mem;
        lif_step(pv_v[j], pv_ad[j],
                 pv_g_exc[j] + ffwd[j], pv_g_inh[j],
                 ws[WS_GAP + j],
                 pv_gL[j], pv_tau[j], pv_vth[j], spk, mem);
        out[j] = spk; out[7500 + j] = mem;
    } else if (i < PV + SST) {
        const int j = i - PV;
        float spk, mem;
        lif_step(sst_v[j], sst_ad[j], sst_g_exc[j], sst_g_inh[j], 0.0f,
                 sst_gL[j], sst_tau[j], sst_vth[j], spk, mem);
        out[3000 + j] = spk; out[10500 + j] = mem;
    } else if (i < PV + SST + VIP) {
        const int j = i - PV - SST;
        const float nic = nic_tr[j] * 0.99004983374916811f + a;  // exp(-DT/50)
        float spk, mem;
        lif_step(vip_v[j], vip_ad[j],
                 vip_g_exc[j] + ws[WS_VIP + j] + 0.25f * nic, vip_g_inh[j],
                 0.0f, vip_gL[j], vip_tau[j], vip_vth[j], spk, mem);
        out[5250 + j] = spk; out[12750 + j] = mem;
    } else if (i < PV + SST + VIP + NGC) {
        const int j = i - PV - SST - VIP;
        const float mus = mus_tr[j] * 0.99750312239746010f + a;  // exp(-DT/200)
        float spk, mem;
        lif_step(ngc_v[j], ngc_ad[j],
                 ngc_g_exc[j] + ws[WS_NGC + j] + 0.2f * mus, ngc_g_inh[j],
                 0.0f, ngc_gL[j], ngc_tau[j], ngc_vth[j], spk, mem);
        out[6750 + j] = spk; out[14250 + j] = mem;
    }
}

// ---------------------------------------------------------------------------
extern "C" void kernel_launch(void* const* d_in, const int* in_sizes, int n_in,
                              void* d_out, int out_size, void* d_ws, size_t ws_size,
                              hipStream_t stream)
{
    const float* pv_g_exc   = (const float*)d_in[0];
    const float* pv_g_inh   = (const float*)d_in[1];
    const float* sst_g_exc  = (const float*)d_in[2];
    const float* sst_g_inh  = (const float*)d_in[3];
    const float* vip_g_exc  = (const float*)d_in[4];
    const float* vip_g_inh  = (const float*)d_in[5];
    const float* ngc_g_exc  = (const float*)d_in[6];
    const float* ngc_g_inh  = (const float*)d_in[7];
    const float* ffwd       = (const float*)d_in[8];
    const float* xlr        = (const float*)d_in[9];
    const float* ach        = (const float*)d_in[10];
    const float* w_lr_vip   = (const float*)d_in[11];
    const float* w_lr_ngc   = (const float*)d_in[12];
    const float* w_pv_gap   = (const float*)d_in[13];
    const float* prev_pv    = (const float*)d_in[14];
    const float* nic_tr     = (const float*)d_in[15];
    const float* mus_tr     = (const float*)d_in[16];
    const float* pv_gL   = (const float*)d_in[17];
    const float* pv_tau  = (const float*)d_in[18];
    const float* pv_vth  = (const float*)d_in[19];
    const float* pv_v    = (const float*)d_in[21];
    const float* pv_ad   = (const float*)d_in[22];
    const float* sst_gL  = (const float*)d_in[23];
    const float* sst_tau = (const float*)d_in[24];
    const float* sst_vth = (const float*)d_in[25];
    const float* sst_v   = (const float*)d_in[27];
    const float* sst_ad  = (const float*)d_in[28];
    const float* vip_gL  = (const float*)d_in[29];
    const float* vip_tau = (const float*)d_in[30];
    const float* vip_vth = (const float*)d_in[31];
    const float* vip_v   = (const float*)d_in[33];
    const float* vip_ad  = (const float*)d_in[34];
    const float* ngc_gL  = (const float*)d_in[35];
    const float* ngc_tau = (const float*)d_in[36];
    const float* ngc_vth = (const float*)d_in[37];
    const float* ngc_v   = (const float*)d_in[39];
    const float* ngc_ad  = (const float*)d_in[40];

    float* ws  = (float*)d_ws;
    float* out = (float*)d_out;

    const int vipTiles = (VIP + 15) / 16;   // 94
    const int ngcTiles = (NGC + 15) / 16;   // 47
    const int pvTiles  = (PV  + 15) / 16;   // 188
    const int nBlocks  = vipTiles + ngcTiles + pvTiles;  // 329

    gemv_wmma_kernel<<<nBlocks, 256, 0, stream>>>(
        w_lr_vip, w_lr_ngc, w_pv_gap, xlr, prev_pv, ws, vipTiles, ngcTiles);

    const int nNeurons = PV + SST + VIP + NGC;  // 7500
    lif_kernel<<<(nNeurons + 255) / 256, 256, 0, stream>>>(
        pv_g_exc, pv_g_inh, sst_g_exc, sst_g_inh,
        vip_g_exc, vip_g_inh, ngc_g_exc, ngc_g_inh,
        ffwd, ach, nic_tr, mus_tr,
        pv_gL, pv_tau, pv_vth, pv_v, pv_ad,
        sst_gL, sst_tau, sst_vth, sst_v, sst_ad,
        vip_gL, vip_tau, vip_vth, vip_v, vip_ad,
        ngc_gL, ngc_tau, ngc_vth, ngc_v, ngc_ad,
        ws, out);
}